// GCN_layer_67740224192671
// MI455X (gfx1250) — compile-verified
//
#include <hip/hip_runtime.h>
#include <math.h>

#define N_NODES 50000
#define N_EDGES 800000
#define D_FEAT  96
#define HIDDEN  96

typedef __attribute__((ext_vector_type(2))) float v2f;
typedef __attribute__((ext_vector_type(8))) float v8f;

// ---------------------------------------------------------------------------
// Zero the segment-sum accumulator (pre) region of d_out.
// ---------------------------------------------------------------------------
__global__ void gcn_zero_kernel(float* __restrict__ p, int n) {
    int i = blockIdx.x * blockDim.x + threadIdx.x;
    if (i < n) p[i] = 0.0f;
}

// ---------------------------------------------------------------------------
// h = x @ W + b  via V_WMMA_F32_16X16X4_F32.
// One wave per 16-row tile of x; 6 column tiles (HIDDEN=96) held as 6 v8f
// accumulators; K loop of 24 steps (K=4 per WMMA). W staged in LDS (36 KB),
// shared by all 8 waves of the block.
// ---------------------------------------------------------------------------
__global__ void __launch_bounds__(256)
gcn_gemm_wmma_kernel(const float* __restrict__ x,
                     const float* __restrict__ W,
                     const float* __restrict__ b,
                     float* __restrict__ h) {
    __shared__ float lW[D_FEAT * HIDDEN];   // 36864 bytes
    for (int i = threadIdx.x; i < D_FEAT * HIDDEN; i += blockDim.x)
        lW[i] = W[i];
    __syncthreads();

    const int lane = threadIdx.x & 31;
    const int wave = threadIdx.x >> 5;
    const int tile = blockIdx.x * 8 + wave;        // 16-row tile index
    if (tile >= (N_NODES / 16)) return;            // wave-uniform guard
    const int row0 = tile * 16;

    const int m  = lane & 15;        // A: row-in-tile; B: col-in-tile
    const int kk = (lane >> 4) * 2;  // K sub-offset per half-wave: 0 or 2

    v8f acc[6];
#pragma unroll
    for (int t = 0; t < 6; ++t) acc[t] = (v8f){0,0,0,0,0,0,0,0};

    for (int k0 = 0; k0 < D_FEAT; k0 += 4) {
        // A fragment (16x4 f32): x[row0+m][k0+kk .. k0+kk+1]
        const float* xp = x + (size_t)(row0 + m) * D_FEAT + k0 + kk;
        v2f a;
        a.x = xp[0];
        a.y = xp[1];
#pragma unroll
        for (int t = 0; t < 6; ++t) {
            // B fragment (4x16 f32): W[k0+kk][t*16+m], W[k0+kk+1][t*16+m]
            v2f bb;
            bb.x = lW[(k0 + kk)     * HIDDEN + t * 16 + m];
            bb.y = lW[(k0 + kk + 1) * HIDDEN + t * 16 + m];
            acc[t] = __builtin_amdgcn_wmma_f32_16x16x4_f32(
                false, a, false, bb, (short)0, acc[t], false, false);
        }
    }

    // D layout: VGPR i -> row = i + 8*(lane>=16), col = lane&15
    const int half = lane >> 4;
#pragma unroll
    for (int t = 0; t < 6; ++t) {
        const int col  = t * 16 + m;
        const float bias = b[col];
#pragma unroll
        for (int i = 0; i < 8; ++i) {
            const int row = row0 + i + half * 8;
            h[(size_t)row * HIDDEN + col] = acc[t][i] + bias;
        }
    }
}

// ---------------------------------------------------------------------------
// pre[r] += w_e * h[c]  — one wave32 per edge, 3 features per lane
// (f = lane, lane+32, lane+64): coalesced 128B gather segments + contiguous
// atomic adds. h and pre both live in L2 (19.2 MB each vs 192 MB).
// ---------------------------------------------------------------------------
__global__ void __launch_bounds__(256)
gcn_scatter_kernel(const int* __restrict__ erow,
                   const int* __restrict__ ecol,
                   const float* __restrict__ ew,
                   const float* __restrict__ h,
                   float* __restrict__ pre) {
    const int lane = threadIdx.x & 31;
    const int wave = threadIdx.x >> 5;
    const int e = blockIdx.x * 8 + wave;
    if (e >= N_EDGES) return;                      // wave-uniform guard

    const int   r  = erow[e];
    const int   c  = ecol[e];
    const float we = ew[e];
    const float* hs = h   + (size_t)c * HIDDEN;
    float*       ps = pre + (size_t)r * HIDDEN;
#pragma unroll
    for (int j = 0; j < 3; ++j) {
        const int f = lane + j * 32;
        atomicAdd(&ps[f], we * hs[f]);
    }
}

// ---------------------------------------------------------------------------
// output = elu(pre)  (alpha = 1)
// ---------------------------------------------------------------------------
__global__ void gcn_elu_kernel(const float* __restrict__ pre,
                               float* __restrict__ out, int n) {
    int i = blockIdx.x * blockDim.x + threadIdx.x;
    if (i < n) {
        float v = pre[i];
        out[i] = v > 0.0f ? v : expm1f(v);
    }
}

// ---------------------------------------------------------------------------
extern "C" void kernel_launch(void* const* d_in, const int* in_sizes, int n_in,
                              void* d_out, int out_size, void* d_ws, size_t ws_size,
                              hipStream_t stream) {
    const float* x  = (const float*)d_in[0];              // [N, 96]
    const float* W  = (const float*)d_in[1];              // [96, 96]
    const float* b  = (const float*)d_in[2];              // [96]
    const int*   ei = (const int*)d_in[3];                // [2, E]: row, col
    const float* ew = (const float*)d_in[4];              // [E]

    const int nh = N_NODES * HIDDEN;
    float* pre = (float*)d_out;                           // output[0]: pre_nonlinearity
    float* out = (float*)d_out + (size_t)nh;              // output[1]: elu; holds h first

    // 1) pre = 0
    gcn_zero_kernel<<<(nh + 255) / 256, 256, 0, stream>>>(pre, nh);

    // 2) h = x @ W + b  (WMMA f32), stored in the output[1] slot
    const int ntiles = N_NODES / 16;                      // 3125, exact
    gcn_gemm_wmma_kernel<<<(ntiles + 7) / 8, 256, 0, stream>>>(x, W, b, out);

    // 3) pre[r] += w_e * h[c]
    gcn_scatter_kernel<<<(N_EDGES + 7) / 8, 256, 0, stream>>>(
        ei, ei + N_EDGES, ew, out, pre);

    // 4) output[1] = elu(pre)   (overwrites the h staging, after scatter done)
    gcn_elu_kernel<<<(nh + 255) / 256, 256, 0, stream>>>(pre, out, nh);
}